// TypePositionalEncoding_77446850281785
// MI455X (gfx1250) — compile-verified
//
#include <hip/hip_runtime.h>
#include <hip/hip_bf16.h>
#include <stdint.h>

#define D_MODEL   512
#define VOCAB     2048
#define MAX_LEN   40000
#define ROWS_PER_BLOCK 8   // one row per wave32, 8 waves per block

typedef float v4f __attribute__((ext_vector_type(4)));

// ---------------- Phase 1: first_seen[v] = INT_MAX ----------------
__global__ void tpe_init_first_seen(int* __restrict__ first_seen) {
    int v = blockIdx.x * blockDim.x + threadIdx.x;
    if (v < VOCAB) first_seen[v] = 0x7fffffff;
}

// ---------------- Phase 2: first_seen[tok[i]] = min(i) ----------------
__global__ void tpe_first_seen(const int* __restrict__ tok,
                               int* __restrict__ first_seen, int L) {
    int i = blockIdx.x * blockDim.x + threadIdx.x;
    if (i < L) atomicMin(&first_seen[tok[i]], i);
}

// ---------------- Phase 3: rank each type by first occurrence ----------------
// pos_of_type[v] = #{ w : first_seen[w] < first_seen[v] }  (clamped to MAX_LEN-1)
// Appearing tokens have distinct first_seen, so this equals cumsum(is_first)-1
// propagated to repeats, exactly as in the reference.
__global__ void tpe_rank(const int* __restrict__ first_seen,
                         int* __restrict__ pos_of_type) {
    int v = blockIdx.x * blockDim.x + threadIdx.x;
    if (v >= VOCAB) return;
    int fs = first_seen[v];
    int r = 0;
#pragma unroll 8
    for (int w = 0; w < VOCAB; ++w)
        r += (first_seen[w] < fs) ? 1 : 0;
    pos_of_type[v] = (r < MAX_LEN - 1) ? r : (MAX_LEN - 1);
}

// ---------------- Phase 4: out = x + pe[pos_of_type[tok[row]]] ----------------
// One 512-float row per wave32. Each wave async-copies its 2KB pe row into a
// private LDS slice (global_load_async_to_lds_b128, IOFFSET applies to both
// LDS and global address), overlaps with NT b128 loads of x, waits ASYNCcnt,
// adds, NT-stores. No cross-wave LDS sharing -> no barrier needed.
__global__ void __launch_bounds__(32 * ROWS_PER_BLOCK)
tpe_add_pe(const int*   __restrict__ tok,
           const float* __restrict__ x,
           const float* __restrict__ pe,
           const int*   __restrict__ pos_of_type,
           float*       __restrict__ out, int L) {
    __shared__ float smem[ROWS_PER_BLOCK * D_MODEL];   // 16 KB / block

    const int wave = threadIdx.x >> 5;
    const int lane = threadIdx.x & 31;
    const int row  = blockIdx.x * ROWS_PER_BLOCK + wave;
    if (row >= L) return;                              // wave-uniform exit

    const int p = pos_of_type[tok[row]];               // gathered pe row index
    const float* pe_row = pe  + (size_t)p   * D_MODEL;
    const float* x_row  = x   + (size_t)row * D_MODEL;
    float*       o_row  = out + (size_t)row * D_MODEL;

    // Per-lane bases: lane covers 4 floats (16B), strided by 128 floats (512B).
    float* lds_elem = &smem[wave * D_MODEL + lane * 4];
    const uint32_t lds_base = (uint32_t)(uintptr_t)lds_elem;       // LDS byte offset
    const uint64_t gbase    = (uint64_t)(uintptr_t)(pe_row + lane * 4);

    // Kick off the async pe-row gather into LDS (ASYNCcnt tracked).
    asm volatile(
        "global_load_async_to_lds_b128 %0, %1, off\n\t"
        "global_load_async_to_lds_b128 %0, %1, off offset:512\n\t"
        "global_load_async_to_lds_b128 %0, %1, off offset:1024\n\t"
        "global_load_async_to_lds_b128 %0, %1, off offset:1536"
        :: "v"(lds_base), "v"(gbase) : "memory");

    // Overlap: stream x (read once -> non-temporal) while the DMA runs.
    v4f xv[4];
#pragma unroll
    for (int k = 0; k < 4; ++k)
        xv[k] = __builtin_nontemporal_load(
            reinterpret_cast<const v4f*>(x_row + lane * 4 + k * 128));

    // Wait for this wave's async LDS writes to land.
    asm volatile("s_wait_asynccnt 0x0" ::: "memory");

#pragma unroll
    for (int k = 0; k < 4; ++k) {
        v4f pv = *reinterpret_cast<const v4f*>(
            &smem[wave * D_MODEL + lane * 4 + k * 128]);
        v4f r = xv[k] + pv;
        __builtin_nontemporal_store(
            r, reinterpret_cast<v4f*>(o_row + lane * 4 + k * 128));
    }
}

extern "C" void kernel_launch(void* const* d_in, const int* in_sizes, int n_in,
                              void* d_out, int out_size, void* d_ws, size_t ws_size,
                              hipStream_t stream) {
    const int*   tok = (const int*)  d_in[0];   // src_token_ids (1, L)
    const float* x   = (const float*)d_in[1];   // embedded_x   (1, L, 512)
    const float* pe  = (const float*)d_in[2];   // pe           (1, 40000, 512)
    float*       out = (float*)      d_out;

    const int L = in_sizes[0];

    int* first_seen  = (int*)d_ws;              // VOCAB ints
    int* pos_of_type = first_seen + VOCAB;      // VOCAB ints

    tpe_init_first_seen<<<(VOCAB + 255) / 256, 256, 0, stream>>>(first_seen);
    tpe_first_seen<<<(L + 255) / 256, 256, 0, stream>>>(tok, first_seen, L);
    tpe_rank<<<(VOCAB + 255) / 256, 256, 0, stream>>>(first_seen, pos_of_type);

    const int blocks = (L + ROWS_PER_BLOCK - 1) / ROWS_PER_BLOCK;
    tpe_add_pe<<<blocks, 32 * ROWS_PER_BLOCK, 0, stream>>>(
        tok, x, pe, pos_of_type, out, L);
}